// ThresholdingAutoEncoderTopK_45320494907997
// MI455X (gfx1250) — compile-verified
//
#include <hip/hip_runtime.h>

typedef __attribute__((ext_vector_type(2))) float        v2f;
typedef __attribute__((ext_vector_type(8))) float        v8f;
typedef __attribute__((ext_vector_type(4))) unsigned int u32x4;
typedef __attribute__((ext_vector_type(8))) int          i32x8;
typedef __attribute__((ext_vector_type(4))) int          i32x4;

#define B_ROWS 4096
#define DIM    768
#define FEAT   49152
#define TOPK   64
#define ROWS   32                 // rows per workgroup
#define CHUNK  64                 // feature columns per chunk (4 col subtiles x 2 row halves)
#define KTILE  128                // K rows staged per pipeline stage
#define KSUB   16                 // K rows staged per wave per TDM op (8 waves x 16 = 128)
#define NSTAGE (DIM / KTILE)      // 6
#define NCHUNK (FEAT / CHUNK)     // 768
#define NT     (NCHUNK * NSTAGE)  // 4608 flat pipeline stages

// ---------------------------------------------------------------------------
// TDM: DMA a KSUB x CHUNK fp32 tile of W (row-major, row stride FEAT) into LDS.
// D# group0: count=1 | lds_addr | global_addr | type=2
// D# group1: data_size=4B, tensor_dim0=FEAT, tensor_dim1=DIM,
//            tile_dim0=CHUNK, tile_dim1=KSUB, tensor_dim0_stride=FEAT
// ---------------------------------------------------------------------------
__device__ __forceinline__ void tdm_load_tile(unsigned long long gaddr, unsigned lds_addr)
{
    u32x4 g0;
    g0[0] = 1u;                                                     // count=1, user mode
    g0[1] = lds_addr;                                               // LDS byte address
    g0[2] = (unsigned)(gaddr & 0xFFFFFFFFull);                      // global addr [31:0]
    g0[3] = ((unsigned)(gaddr >> 32) & 0x01FFFFFFu) | (2u << 30);   // addr [56:32] | type=2
    i32x8 g1;
    g1[0] = 0x00020000;                                             // data_size=4B, no multicast
    g1[1] = (int)(((unsigned)FEAT & 0xFFFFu) << 16);                // tensor_dim0 lo16
    g1[2] = (int)((((unsigned)FEAT >> 16) & 0xFFFFu)
                | (((unsigned)DIM & 0xFFFFu) << 16));               // dim0 hi16 | dim1 lo16
    g1[3] = (int)((((unsigned)DIM >> 16) & 0xFFFFu)
                | ((unsigned)CHUNK << 16));                         // dim1 hi16 | tile_dim0
    g1[4] = KSUB;                                                   // tile_dim1 | tile_dim2=0
    g1[5] = FEAT;                                                   // tensor_dim0_stride lo32
    g1[6] = 0;                                                      // stride hi | dim1_stride lo
    g1[7] = 0;
    i32x4 gz4 = {0, 0, 0, 0};
    i32x8 gz8 = {0, 0, 0, 0, 0, 0, 0, 0};
    __builtin_amdgcn_tensor_load_to_lds(g0, g1, gz4, gz4, gz8, 0);
}

// ---------------------------------------------------------------------------
// Kernel 1: fused encode (fp32 WMMA, TDM-staged B operand) + per-row top-64.
// LDS layout puts the W double-buffer at offset 0 so all B-fragment DS loads
// use 16-bit immediate offsets (<= 32 KB) off one per-lane base VGPR.
// ---------------------------------------------------------------------------
__global__ __launch_bounds__(256) void
sae_encode_topk(const float* __restrict__ x,
                const float* __restrict__ W,      // [DIM, FEAT] row-major
                const float* __restrict__ b_dec,
                float* __restrict__ out_vals,     // [B_ROWS, TOPK]
                int*   __restrict__ out_cols)     // [B_ROWS, TOPK]
{
    extern __shared__ char smem_raw[];
    float* wbuf  = (float*)smem_raw;                 // 2*KTILE*CHUNK (64 KB, offset 0)
    float* xs    = wbuf + 2 * KTILE * CHUNK;         // ROWS*DIM  (x - b_dec)
    float* topv  = xs + ROWS * DIM;                  // ROWS*TOPK
    int*   topc  = (int*)(topv + ROWS * TOPK);       // ROWS*TOPK
    float* candv = (float*)(topc + ROWS * TOPK);     // ROWS*CHUNK
    int*   candc = (int*)(candv + ROWS * CHUNK);     // ROWS*CHUNK
    int*   candn = candc + ROWS * CHUNK;             // ROWS
    float* minv  = (float*)(candn + ROWS);           // ROWS
    int*   minp  = (int*)(minv + ROWS);              // ROWS

    const int tid  = threadIdx.x;
    const int lane = tid & 31;
    const int wave = tid >> 5;
    const int row0 = blockIdx.x * ROWS;

    for (int i = tid; i < ROWS * DIM; i += 256) {
        int r = i / DIM, d = i - r * DIM;
        xs[i] = x[(size_t)(row0 + r) * DIM + d] - b_dec[d];
    }
    for (int i = tid; i < ROWS * TOPK; i += 256) { topv[i] = 0.0f; topc[i] = 0; }
    if (tid < ROWS) { candn[tid] = 0; minv[tid] = 0.0f; minp[tid] = 0; }
    __syncthreads();

    // SGPR-uniform values for the TDM descriptor.
    const int wv = __builtin_amdgcn_readfirstlane(wave);
    const unsigned long long wbase   = (unsigned long long)(uintptr_t)W;
    const unsigned           ldsbase = (unsigned)(uintptr_t)wbuf;

    auto issue = [&](int t) {
        int chunk = t / NSTAGE, stage = t - chunk * NSTAGE;
        int k0 = stage * KTILE, c0 = chunk * CHUNK;
        unsigned long long g = wbase +
            4ull * ((unsigned long long)(unsigned)(k0 + wv * KSUB) * FEAT + (unsigned)c0);
        unsigned l = ldsbase + (unsigned)(((t & 1) * KTILE + wv * KSUB) * CHUNK * 4);
        tdm_load_tile(g, l);
    };

    const int rbase   = (wave >> 2) * 16;   // 0 or 16
    const int csub    = (wave & 3) * 16;    // column subtile within chunk
    const int mrow    = lane & 15;
    const int koff    = (lane >> 4) * 2;    // A/B: lanes 16-31 hold K=2,3
    const int crow_hi = (lane >> 4) * 8;    // C: lanes 16-31 hold M=v+8
    const float* xrow = xs + (size_t)(rbase + mrow) * DIM;
    const v8f vzero = {};

    v8f acc = {};
    issue(0);
    for (int t = 0; t < NT; ++t) {
        const int chunk = t / NSTAGE, stage = t - chunk * NSTAGE;
        if (t + 1 < NT) { issue(t + 1); __builtin_amdgcn_s_wait_tensorcnt(1); }
        else            { __builtin_amdgcn_s_wait_tensorcnt(0); }
        __syncthreads();                      // all 8 waves' stage-t tiles resident

        if (stage == 0) acc = vzero;
        const int    k0 = stage * KTILE;
        // Per-lane B base: buffer select + this lane's column; k offsets become
        // 16-bit DS immediates ((kk+koff)*CHUNK*4 <= 32 KB).
        const float* bl = wbuf + (t & 1) * KTILE * CHUNK + csub + mrow;
        #pragma unroll 4
        for (int kk = 0; kk < KTILE; kk += 4) {
            v2f a = *(const v2f*)(xrow + k0 + kk + koff);
            v2f b;
            b.x = bl[(kk + koff    ) * CHUNK];
            b.y = bl[(kk + koff + 1) * CHUNK];
            acc = __builtin_amdgcn_wmma_f32_16x16x4_f32(
                false, a, false, b, (short)0, acc, false, false);
        }

        if (stage == NSTAGE - 1) {            // full K accumulated: run top-k filter
            const int col = chunk * CHUNK + csub + mrow;
            #pragma unroll
            for (int v = 0; v < 8; ++v) {
                float val = acc[v];
                int   r   = rbase + crow_hi + v;
                if (__builtin_fabsf(val) > minv[r]) {
                    int p = atomicAdd(&candn[r], 1);
                    if (p < CHUNK) { candv[r * CHUNK + p] = val; candc[r * CHUNK + p] = col; }
                }
            }
            __syncthreads();
            if (tid < ROWS) {
                int   r  = tid;
                float mv = minv[r];
                int   mp = minp[r];
                int   n  = candn[r]; n = n > CHUNK ? CHUNK : n;
                for (int j = 0; j < n; ++j) {
                    float cv = candv[r * CHUNK + j];
                    if (__builtin_fabsf(cv) > mv) {
                        topv[r * TOPK + mp] = cv;
                        topc[r * TOPK + mp] = candc[r * CHUNK + j];
                        mv = __builtin_fabsf(topv[r * TOPK]); mp = 0;
                        for (int u = 1; u < TOPK; ++u) {
                            float av = __builtin_fabsf(topv[r * TOPK + u]);
                            if (av < mv) { mv = av; mp = u; }
                        }
                    }
                }
                minv[r] = mv; minp[r] = mp; candn[r] = 0;
            }
        }
        __syncthreads();                      // buffer (t&1) reusable at t+2
    }

    for (int i = tid; i < ROWS * TOPK; i += 256) {
        int r = i / TOPK, j = i - r * TOPK;
        out_vals[(size_t)(row0 + r) * TOPK + j] = topv[i];
        out_cols[(size_t)(row0 + r) * TOPK + j] = topc[i];
    }
}

// ---------------------------------------------------------------------------
// Kernel 2: sparse decode  x_hat[b, d] = b_dec[d] + sum_k val[b,k]*W[d, col[b,k]]
// ---------------------------------------------------------------------------
__global__ __launch_bounds__(256) void
sae_decode(const float* __restrict__ W,
           const float* __restrict__ b_dec,
           const float* __restrict__ vals,
           const int*   __restrict__ cols,
           float*       __restrict__ out)
{
    __shared__ float sv[TOPK];
    __shared__ int   sc[TOPK];
    const int row = blockIdx.y;
    const int d   = blockIdx.x * 256 + threadIdx.x;
    if (threadIdx.x < TOPK) {
        sv[threadIdx.x] = vals[(size_t)row * TOPK + threadIdx.x];
        sc[threadIdx.x] = cols[(size_t)row * TOPK + threadIdx.x];
    }
    __syncthreads();
    float acc = b_dec[d];
    const float* wrow = W + (size_t)d * FEAT;
    #pragma unroll 8
    for (int k = 0; k < TOPK; ++k)
        acc = fmaf(sv[k], wrow[sc[k]], acc);
    out[(size_t)row * DIM + d] = acc;
}

// ---------------------------------------------------------------------------
extern "C" void kernel_launch(void* const* d_in, const int* in_sizes, int n_in,
                              void* d_out, int out_size, void* d_ws, size_t ws_size,
                              hipStream_t stream) {
    (void)in_sizes; (void)n_in; (void)out_size; (void)ws_size;
    const float* x = (const float*)d_in[0];   // [4096, 768]
    const float* W = (const float*)d_in[1];   // [768, 49152]
    const float* b = (const float*)d_in[2];   // [768]
    // d_in[3] is k (== 64), hardcoded as TOPK.

    float* vals = (float*)d_ws;                          // [4096, 64]
    int*   cols = (int*)(vals + (size_t)B_ROWS * TOPK);  // [4096, 64]
    float* out  = (float*)d_out;

    const size_t lds_bytes =
        sizeof(float) * 2 * KTILE * CHUNK +                // wbuf double buffer (offset 0)
        sizeof(float) * ROWS * DIM +                       // xs
        (sizeof(float) + sizeof(int)) * ROWS * TOPK +      // topv/topc
        (sizeof(float) + sizeof(int)) * ROWS * CHUNK +     // candv/candc
        ROWS * (sizeof(int) + sizeof(float) + sizeof(int));

    sae_encode_topk<<<B_ROWS / ROWS, 256, lds_bytes, stream>>>(x, W, b, vals, cols);
    sae_decode<<<dim3(DIM / 256, B_ROWS), 256, 0, stream>>>(W, b, vals, cols, out);
}